// Model_74783970558047
// MI455X (gfx1250) — compile-verified
//
#include <hip/hip_runtime.h>
#include <hip/hip_bf16.h>

typedef __attribute__((ext_vector_type(2))) float v2f;
typedef __attribute__((ext_vector_type(8))) float v8f;

#define KCLUST 1024
#define DIM    128

// ---------------------------------------------------------------- zero scratch
__global__ void zero_kernel(float* p, int n) {
    int i = blockIdx.x * blockDim.x + threadIdx.x;
    if (i < n) p[i] = 0.0f;
}

// ------------------------------------------------- segment sums (wave / point)
// one wave32 per point: lane L accumulates dims [4L, 4L+4)
// unsafeAtomicAdd -> native GLOBAL_ATOMIC_ADD_F32 (no-return, STOREcnt path)
__global__ void accum_kernel(const float* __restrict__ vec,
                             const int* __restrict__ assign,
                             float* __restrict__ sums,
                             float* __restrict__ counts, int nPoints) {
    int gid   = blockIdx.x * blockDim.x + threadIdx.x;
    int point = gid >> 5;
    int lane  = gid & 31;
    if (point >= nPoints) return;
    // load the (wave-uniform) cluster id once, broadcast to all lanes
    int a0 = 0;
    if (lane == 0) a0 = assign[point];
    const int a = __shfl(a0, 0, 32);

    const float4 v = *(const float4*)(vec + (size_t)point * DIM + lane * 4);
    float* s = sums + (size_t)a * DIM + lane * 4;
    unsafeAtomicAdd(s + 0, v.x);
    unsafeAtomicAdd(s + 1, v.y);
    unsafeAtomicAdd(s + 2, v.z);
    unsafeAtomicAdd(s + 3, v.w);
    if (lane == 0) unsafeAtomicAdd(counts + a, 1.0f);
}

// ------------------------------------ centroid = sum/count ; c2 = ||centroid||^2
__global__ void centroid_kernel(const float* __restrict__ sums,
                                const float* __restrict__ counts,
                                float* __restrict__ centroids,
                                float* __restrict__ c2) {
    __shared__ float red[DIM];
    int k = blockIdx.x;
    int d = threadIdx.x;
    float cnt = counts[k];
    float c = sums[(size_t)k * DIM + d] / cnt;
    centroids[(size_t)k * DIM + d] = c;
    red[d] = c * c;
    __syncthreads();
    for (int s = DIM / 2; s > 0; s >>= 1) {
        if (d < s) red[d] += red[d + s];
        __syncthreads();
    }
    if (d == 0) c2[k] = red[0];
}

// ------------------------------------------------------- WMMA argmin kernel
// One wave per 16-point tile. A = points (16xD, fp32), B = centroid tile
// (Dx16, staged in LDS with pad-132 stride), C/D = 16x16 fp32 accumulators.
// dist = c2[k] - 2*dot(c,x)  (x2 is constant per point -> irrelevant to argmin)
#define LDS_STRIDE 132
__global__ void assign_kernel(const float* __restrict__ vec,
                              const float* __restrict__ cent,
                              const float* __restrict__ c2,
                              float* __restrict__ outAssign, int nPoints) {
    __shared__ float ldsB[16 * LDS_STRIDE];

    const int lane      = threadIdx.x & 31;
    const int waveInBlk = threadIdx.x >> 5;
    const int nTiles    = nPoints / 16;          // N is a multiple of 16
    int tile  = blockIdx.x * 8 + waveInBlk;
    bool valid = (tile < nTiles);
    int tclamp = valid ? tile : (nTiles - 1);
    const int basePt  = tclamp * 16;
    const int halfSel = lane >> 4;               // 0: lanes 0-15, 1: lanes 16-31
    const int n       = lane & 15;               // column (centroid within tile)

    // ---- preload the whole A tile row for this lane: 32 steps x float2
    // A 16x4 layout: lanes 0-15 hold K = d+0,d+1 ; lanes 16-31 hold K = d+2,d+3
    v2f a[32];
    const float* vrow = vec + (size_t)(basePt + n) * DIM + (halfSel << 1);
    #pragma unroll
    for (int j = 0; j < 32; ++j)
        a[j] = *(const v2f*)(vrow + j * 4);

    float bestVal[8];
    int   bestIdx[8];
    #pragma unroll
    for (int r = 0; r < 8; ++r) { bestVal[r] = 3.4e38f; bestIdx[r] = 0; }

    const int kb = halfSel;                      // B row for b.x ; b.y = +2

    for (int ct = 0; ct < KCLUST / 16; ++ct) {
        // ---- cooperative stage of 16x128 centroid tile into padded LDS
        {
            int t   = threadIdx.x;
            int row = t >> 4;                    // 0..15
            int col = (t & 15) * 8;              // 0..120
            const float* src = cent + (size_t)(ct * 16 + row) * DIM + col;
            float4 p0 = *(const float4*)(src);
            float4 p1 = *(const float4*)(src + 4);
            float* dst = &ldsB[row * LDS_STRIDE + col];
            *(float4*)(dst)     = p0;
            *(float4*)(dst + 4) = p1;
        }
        __syncthreads();

        v8f acc = {};
        #pragma unroll
        for (int j = 0; j < 32; ++j) {
            const int d = j * 4;
            v2f b;
            b.x = ldsB[n * LDS_STRIDE + d + kb];
            b.y = ldsB[n * LDS_STRIDE + d + kb + 2];
            acc = __builtin_amdgcn_wmma_f32_16x16x4_f32(
                false, a[j], false, b, (short)0, acc, false, false);
        }

        const float c2l = c2[ct * 16 + n];
        const int   idx = ct * 16 + n;
        #pragma unroll
        for (int r = 0; r < 8; ++r) {
            float dist = c2l - 2.0f * acc[r];
            if (dist < bestVal[r] ||
                (dist == bestVal[r] && idx < bestIdx[r])) {
                bestVal[r] = dist;
                bestIdx[r] = idx;
            }
        }
        __syncthreads();
    }

    // ---- butterfly argmin across the 16 lanes of each half-wave
    #pragma unroll
    for (int r = 0; r < 8; ++r) {
        float v = bestVal[r];
        int   i = bestIdx[r];
        #pragma unroll
        for (int m2 = 8; m2 >= 1; m2 >>= 1) {
            float ov = __shfl_xor(v, m2, 32);
            int   oi = __shfl_xor(i, m2, 32);
            if (ov < v || (ov == v && oi < i)) { v = ov; i = oi; }
        }
        if (valid && n == 0) {
            int pt = basePt + (halfSel ? (8 + r) : r);
            outAssign[pt] = (float)i;
        }
    }
}

// ------------------------------------------------------------------ launcher
extern "C" void kernel_launch(void* const* d_in, const int* in_sizes, int n_in,
                              void* d_out, int out_size, void* d_ws, size_t ws_size,
                              hipStream_t stream) {
    const float* vectors    = (const float*)d_in[0];
    const int*   assignment = (const int*)d_in[1];
    const int N = in_sizes[1];                 // 500000

    float* out       = (float*)d_out;
    float* centroids = out;                    // K*D floats
    float* outAssign = out + KCLUST * DIM;     // N values (indices as float)

    float* sums   = (float*)d_ws;              // K*D
    float* counts = sums + KCLUST * DIM;       // K
    float* c2     = counts + KCLUST;           // K

    // 1) zero scratch accumulators
    {
        int n0 = KCLUST * DIM + KCLUST;
        zero_kernel<<<(n0 + 255) / 256, 256, 0, stream>>>(sums, n0);
    }
    // 2) segment sums + counts
    {
        long long threads = (long long)N * 32;
        int blocks = (int)((threads + 255) / 256);
        accum_kernel<<<blocks, 256, 0, stream>>>(vectors, assignment, sums, counts, N);
    }
    // 3) centroids + squared norms
    centroid_kernel<<<KCLUST, DIM, 0, stream>>>(sums, counts, centroids, c2);
    // 4) WMMA distance + argmin
    {
        int tiles  = (N + 15) / 16;
        int blocks = (tiles + 7) / 8;
        assign_kernel<<<blocks, 256, 0, stream>>>(vectors, centroids, c2, outAssign, N);
    }
}